// BSA_42545946034971
// MI455X (gfx1250) — compile-verified
//
#include <hip/hip_runtime.h>
#include <hip/hip_bf16.h>
#include <stdint.h>

// ---------------- problem constants (from reference) ----------------
#define B_ROWS   8192
#define T_LEN    2048
#define F_LEN    20
#define N_STEPS  (T_LEN - F_LEN)          // 2028
#define BSA_THRESHOLD 0.9952f

// ---------------- tiling ----------------
#define CH        64                      // columns per TDM tile / chunk
#define NCH       (T_LEN / CH)            // 32 chunks, exact
#define RPW       32                      // rows per wave (one row per lane)
#define LSTRIDE   (CH + 1)                // padded LDS row stride (dwords) = 65

#if __has_builtin(__builtin_amdgcn_tensor_load_to_lds)
#define HAVE_TDM 1
#else
#define HAVE_TDM 0
#endif

typedef unsigned int u32;
typedef u32 u32x4 __attribute__((ext_vector_type(4)));
typedef int  i32x4 __attribute__((ext_vector_type(4)));
typedef int  i32x8 __attribute__((ext_vector_type(8)));

// Low 32 bits of a flat shared-aperture address == workgroup-relative LDS byte offset.
__device__ __forceinline__ u32 lds_off(const void* p) {
  return (u32)(uintptr_t)p;
}

__device__ __forceinline__ void wait_tensorcnt_le(int n) {
#if __has_builtin(__builtin_amdgcn_s_wait_tensorcnt)
  if (n == 0) __builtin_amdgcn_s_wait_tensorcnt(0);
  else        __builtin_amdgcn_s_wait_tensorcnt(1);
#else
  if (n == 0) asm volatile("s_wait_tensorcnt 0x0" ::: "memory");
  else        asm volatile("s_wait_tensorcnt 0x1" ::: "memory");
#endif
  asm volatile("" ::: "memory");   // keep compiler from hoisting LDS reads above the wait
}

#if HAVE_TDM
// TDM: async 2D strided tile load (RPW rows x CH f32 cols, row stride T_LEN)
// global -> LDS, with hardware padding of 1 DWORD per 64 DWORDs so each LDS
// row has stride 65 dwords (bank-conflict-free lane-per-row access).
__device__ __forceinline__ void tdm_load_tile(u32 lds_byte, const float* gsrc) {
  unsigned long long ga = (unsigned long long)(uintptr_t)gsrc;
  u32x4 g0;
  g0[0] = 1u;                                          // count=1, user descriptor
  g0[1] = lds_byte;                                    // lds_addr
  g0[2] = (u32)ga;                                     // global_addr[31:0]
  g0[3] = ((u32)(ga >> 32) & 0x01FFFFFFu) | (2u << 30);// global_addr[56:32] | type=2
  i32x8 g1;
  g1[0] = (2 << 16)        // data_size = 4B
        | (1 << 20)        // pad_enable
        | (5 << 22);       // pad_interval: every 64 DWORDs
                           // pad_amount field = 0 -> 1 DWORD pad
  g1[1] = (CH  & 0xFFFF) << 16;                        // tensor_dim0 lo16 (=64)
  g1[2] = (RPW & 0xFFFF) << 16;                        // tensor_dim1 lo16 (=32)
  g1[3] = (CH  & 0xFFFF) << 16;                        // tile_dim0 (=64)
  g1[4] = RPW;                                         // tile_dim1 (=32), tile_dim2=0
  g1[5] = T_LEN;                                       // tensor_dim0_stride lo32 (=2048)
  g1[6] = 0;
  g1[7] = 0;
  i32x4 z4 = {0, 0, 0, 0};
#if defined(__clang_major__) && (__clang_major__ >= 23)
  i32x8 z8 = {0, 0, 0, 0, 0, 0, 0, 0};
  __builtin_amdgcn_tensor_load_to_lds(g0, g1, z4, z4, z8, 0);
#else
  __builtin_amdgcn_tensor_load_to_lds(g0, g1, z4, z4, 0);
#endif
}
#else
// Fallback: synchronous coalesced fill of the same padded LDS layout.
__device__ __forceinline__ void fill_tile_sync(float* dst, const float* gsrc, int lane) {
#pragma unroll 4
  for (int r = 0; r < RPW; ++r) {
    dst[r * LSTRIDE + lane]      = gsrc[(size_t)r * T_LEN + lane];
    dst[r * LSTRIDE + 32 + lane] = gsrc[(size_t)r * T_LEN + 32 + lane];
  }
}
#endif

// Balanced-tree sum of the 20-sample window (short dependency chain).
__device__ __forceinline__ float window_sum(const float w[F_LEN]) {
  float a0 = w[0] + w[1],  a1 = w[2] + w[3],  a2 = w[4] + w[5];
  float a3 = w[6] + w[7],  a4 = w[8] + w[9],  a5 = w[10] + w[11];
  float a6 = w[12] + w[13], a7 = w[14] + w[15], a8 = w[16] + w[17];
  float a9 = w[18] + w[19];
  float b0 = a0 + a1, b1 = a2 + a3, b2 = a4 + a5, b3 = a6 + a7, b4 = a8 + a9;
  float c0 = b0 + b1, c1 = b2 + b3;
  return (c0 + c1) + b4;
}

// One BSA step: decide mask from current window d[t..t+F-1], record it, then
// shift window to d'[t+1..t+F] with the conditional filter subtraction applied.
// mf is exactly 0.0 or 1.0 so mf*f[j] is exact (FMA == mul+sub bitwise).
__device__ __forceinline__ void bsa_step(float w[F_LEN], const float f[F_LEN],
                                         float fsum, float enter,
                                         float* lout_row, int col) {
  float s  = window_sum(w);
  float mf = (fabsf(s - fsum) <= fabsf(s) * BSA_THRESHOLD) ? 1.0f : 0.0f;
  lout_row[col] = mf;
#pragma unroll
  for (int j = 0; j < F_LEN - 1; ++j) w[j] = w[j + 1] - mf * f[j];
  w[F_LEN - 1] = enter - mf * f[F_LEN - 1];
}

// Coalesced store of one finished 32x64 mask tile (tile index tt).
__device__ __forceinline__ void flush_tile(float* __restrict__ out, const float* lout,
                                           int lane, int row_base, int tt) {
  const size_t colbase = (size_t)tt * CH;
#pragma unroll 4
  for (int j = 0; j < RPW; ++j) {
    float v0 = lout[j * LSTRIDE + lane];
    float v1 = lout[j * LSTRIDE + 32 + lane];
    size_t o = (size_t)(row_base + j) * T_LEN + colbase;
    out[o + lane]      = v0;
    out[o + 32 + lane] = v1;
  }
}

__global__ __launch_bounds__(RPW) void BSA_42545946034971_kernel(
    const float* __restrict__ sig, const float* __restrict__ filt,
    float* __restrict__ out) {
  __shared__ float lin[2][RPW * LSTRIDE];   // double-buffered input tiles
  __shared__ float lout[RPW * LSTRIDE];     // mask staging tile

  const int lane     = threadIdx.x;                 // 0..31, one row per lane
  const int row_base = blockIdx.x * RPW;
  const float* gbase = sig + (size_t)row_base * T_LEN;

  // Filter taps (uniform address -> scalar loads) + filter sum.
  float f[F_LEN];
  float fsum = 0.0f;
#pragma unroll
  for (int j = 0; j < F_LEN; ++j) { f[j] = filt[j]; fsum += f[j]; }

  // Kick off the first two tiles.
#if HAVE_TDM
  tdm_load_tile(lds_off(&lin[0][0]), gbase);
  tdm_load_tile(lds_off(&lin[1][0]), gbase + CH);
#else
  fill_tile_sync(&lin[0][0], gbase, lane);
  fill_tile_sync(&lin[1][0], gbase + CH, lane);
#endif

  float* lin0_row = &lin[0][lane * LSTRIDE];
  float* lout_row = &lout[lane * LSTRIDE];
  float  w[F_LEN];

  // ---- chunk 0: init window from cols 0..19, steps consume cols 20..63 ----
#if HAVE_TDM
  wait_tensorcnt_le(1);                       // chunk 0 arrived (in-order TDM)
#endif
#pragma unroll
  for (int j = 0; j < F_LEN; ++j) w[j] = lin0_row[j];
#pragma unroll
  for (int k = F_LEN; k < CH; ++k) {
    // step t = k-20, tile 0, col = k-20
    bsa_step(w, f, fsum, lin0_row[k], lout_row, k - F_LEN);
  }

  // ---- chunks 1..31 ----
  for (int c = 1; c < NCH; ++c) {
    const int bsel = c & 1;
    if (c + 1 < NCH) {
      // Buffer (c+1)&1 held chunk c-1, fully consumed last iteration.
      asm volatile("s_wait_dscnt 0x0" ::: "memory");
#if HAVE_TDM
      tdm_load_tile(lds_off(&lin[(c + 1) & 1][0]), gbase + (size_t)(c + 1) * CH);
#else
      fill_tile_sync(&lin[(c + 1) & 1][0], gbase + (size_t)(c + 1) * CH, lane);
#endif
    }
#if HAVE_TDM
    wait_tensorcnt_le((c + 1 < NCH) ? 1 : 0);  // chunk c resident
#endif
    const float* lrow = &lin[bsel][lane * LSTRIDE];
#pragma unroll
    for (int k = 0; k < CH; ++k) {
      // step t = c*64 + k - 20; col = (k + 44) & 63 (static after unroll)
      bsa_step(w, f, fsum, lrow[k], lout_row, (k + (CH - F_LEN)) & (CH - 1));
      if (k == F_LEN - 1) {
        // tile (c-1) is now complete: cols 0..43 from chunk c-1, 44..63 just written
        flush_tile(out, lout, lane, row_base, c - 1);
      }
    }
  }

  // ---- tail: steps 2028..2047 are zero; finish tile 31 ----
#pragma unroll
  for (int col = CH - F_LEN; col < CH; ++col) lout_row[col] = 0.0f;
  flush_tile(out, lout, lane, row_base, NCH - 1);
}

extern "C" void kernel_launch(void* const* d_in, const int* in_sizes, int n_in,
                              void* d_out, int out_size, void* d_ws, size_t ws_size,
                              hipStream_t stream) {
  (void)in_sizes; (void)n_in; (void)out_size; (void)d_ws; (void)ws_size;
  const float* sig  = (const float*)d_in[0];   // [8192, 2048] f32
  const float* filt = (const float*)d_in[1];   // [20] f32
  float* out = (float*)d_out;                  // [8192, 2048] f32
  dim3 grid(B_ROWS / RPW);                     // 256 single-wave workgroups
  dim3 block(RPW);                             // 32 threads = 1 wave32
  BSA_42545946034971_kernel<<<grid, block, 0, stream>>>(sig, filt, out);
}